// Cross_modal_ContrastiveLoss6_91207925497989
// MI455X (gfx1250) — compile-verified
//
#include <hip/hip_runtime.h>

// Problem constants (from reference setup_inputs)
#define N_SAMP 4096
#define DIMS   256
#define NCLS   64
#define MARGIN 0.3f

typedef __attribute__((ext_vector_type(2))) float v2f;
typedef __attribute__((ext_vector_type(8))) float v8f;

// ---------------- workspace layout (bytes) ----------------
// [0,256)              int   counts[64]
// [256, 256+64K)       float sumR[64*256]
// [.. +64K)            float sumT[64*256]
// [.. +256)            float nR[64]
// [.. +256)            float nT[64]
// [.. +256)            float nC[64]
// [.. +4)              float lossAcc
#define WS_WORDS ((256 + 2*NCLS*DIMS*4 + 3*NCLS*4 + 4) / 4)

__global__ void k_zero(unsigned int* ws) {
    for (int i = blockIdx.x * blockDim.x + threadIdx.x; i < WS_WORDS;
         i += gridDim.x * blockDim.x)
        ws[i] = 0u;
}

__global__ void k_counts(const int* __restrict__ targets, int* __restrict__ counts) {
    int i = blockIdx.x * blockDim.x + threadIdx.x;
    if (i < N_SAMP) atomicAdd(&counts[targets[i]], 1);
}

// Dim-sliced class sums: grid = 8 dim-chunks x 16 row-chunks = 128 blocks.
// Each block accumulates [64 classes x 32 dims] x 2 modalities in LDS,
// reading its slice of the inputs exactly once, then flushes with atomics.
__global__ void k_sums(const float* __restrict__ m1, const float* __restrict__ m2,
                       const int* __restrict__ targets,
                       float* __restrict__ sumR, float* __restrict__ sumT) {
    __shared__ float ls[2 * NCLS * 32];  // 16 KB
    const int tid = threadIdx.x;
    for (int i = tid; i < 2 * NCLS * 32; i += 256) ls[i] = 0.f;
    __syncthreads();

    const int dch = blockIdx.x & 7;   // dim chunk: dims [dch*32, dch*32+32)
    const int rch = blockIdx.x >> 3;  // row chunk: rows [rch*256, rch*256+256)
    const int row = rch * 256 + tid;
    const int c = targets[row];
    const float* p1 = m1 + (size_t)row * DIMS + dch * 32;
    const float* p2 = m2 + (size_t)row * DIMS + dch * 32;
#pragma unroll 8
    for (int j = 0; j < 32; ++j) {
        atomicAdd(&ls[c * 32 + j], p1[j]);
        atomicAdd(&ls[2048 + c * 32 + j], p2[j]);
    }
    __syncthreads();

    for (int i = tid; i < 2 * NCLS * 32; i += 256) {
        const int mod = i >> 11;
        const int idx = i & 2047;
        const int cc = idx >> 5, j = idx & 31;
        float v = ls[i];
        float* dst = (mod ? sumT : sumR) + cc * DIMS + dch * 32 + j;
        atomicAdd(dst, v);
    }
}

// Per-class squared norms of cR, cT, cC = 0.5*(cR+cT). grid=64, block=64.
__global__ void k_norms(const int* __restrict__ counts,
                        const float* __restrict__ sumR, const float* __restrict__ sumT,
                        float* __restrict__ nR, float* __restrict__ nT,
                        float* __restrict__ nC) {
    __shared__ float sh[3 * 64];
    const int c = blockIdx.x, t = threadIdx.x;
    const float cnt = (float)counts[c];
    const float inv = cnt > 0.f ? 1.f / cnt : 0.f;
    float aR = 0.f, aT = 0.f, aC = 0.f;
    for (int k = t; k < DIMS; k += 64) {
        float r = sumR[c * DIMS + k] * inv;
        float tt = sumT[c * DIMS + k] * inv;
        float cc = 0.5f * (r + tt);
        aR += r * r; aT += tt * tt; aC += cc * cc;
    }
    sh[t] = aR; sh[64 + t] = aT; sh[128 + t] = aC;
    __syncthreads();
    if (t < 3) {
        float s = 0.f;
        for (int i = 0; i < 64; ++i) s += sh[t * 64 + i];
        (t == 0 ? nR : (t == 1 ? nT : nC))[c] = s;
    }
}

// WMMA stage: 64x64 Gram matrices G1 = cR.cC^T, G2 = cT.cC^T via
// V_WMMA_F32_16X16X4_F32 (full fp32). One wave per 16x16 tile, 16 tiles.
// A-frag 16x4 fp32 layout: lanes 0-15 -> M=lane, K=k0..k0+1; lanes 16-31 -> same M, K=k0+2..k0+3.
// B-frag 4x16 mirrors it with N instead of M. C/D: vgpr r -> M = r + 8*(lane>=16), N = lane%16.
__global__ __launch_bounds__(32) void k_wmma(
    const int* __restrict__ counts,
    const float* __restrict__ sumR, const float* __restrict__ sumT,
    const float* __restrict__ nR, const float* __restrict__ nT,
    const float* __restrict__ nC, float* __restrict__ lossAcc) {
    const int tile = blockIdx.x;
    const int tm = tile >> 2, tn = tile & 3;
    const int lane = threadIdx.x;
    const int half = lane >> 4;
    const int m = lane & 15;
    const int a = tm * 16 + m;  // class index for A row (same for both lane halves)
    const int b = tn * 16 + m;  // class index for B column
    const float cntA = (float)counts[a];
    const float cntB = (float)counts[b];
    const float invA = cntA > 0.f ? 1.f / cntA : 0.f;
    const float invB = cntB > 0.f ? 0.5f / cntB : 0.f;  // folds the 0.5 of cC
    const float* pRa = sumR + a * DIMS;
    const float* pTa = sumT + a * DIMS;
    const float* pRb = sumR + b * DIMS;
    const float* pTb = sumT + b * DIMS;

    v8f accR = {}, accT = {};
    for (int k0 = 0; k0 < DIMS; k0 += 4) {
        const int ks = k0 + 2 * half;
        v2f ar = *(const v2f*)(pRa + ks) * invA;                             // cR row frag
        v2f at = *(const v2f*)(pTa + ks) * invA;                             // cT row frag
        v2f bc = (*(const v2f*)(pRb + ks) + *(const v2f*)(pTb + ks)) * invB; // cC col frag
        accR = __builtin_amdgcn_wmma_f32_16x16x4_f32(false, ar, false, bc,
                                                     (short)0, accR, false, false);
        accT = __builtin_amdgcn_wmma_f32_16x16x4_f32(false, at, false, bc,
                                                     (short)0, accT, false, false);
    }

    const float nCb = nC[b];
    const float invN2 = 1.f / ((float)N_SAMP * (float)N_SAMP);
    float local = 0.f;
#pragma unroll
    for (int r = 0; r < 8; ++r) {
        const int arow = tm * 16 + r + 8 * half;
        const float w = (float)counts[arow] * cntB * invN2;
        // modal R vs centers
        float d2 = nR[arow] + nCb - 2.f * accR[r];
        d2 = fmaxf(d2, 1e-12f);
        float dd = sqrtf(sqrtf(d2) + 1e-10f);
        float h = fmaxf(MARGIN - dd, 0.f);
        float term = (arow == b) ? d2 : h * h;
        // modal T vs centers
        float d2t = nT[arow] + nCb - 2.f * accT[r];
        d2t = fmaxf(d2t, 1e-12f);
        float ddt = sqrtf(sqrtf(d2t) + 1e-10f);
        float ht = fmaxf(MARGIN - ddt, 0.f);
        float termt = (arow == b) ? d2t : ht * ht;
        local += w * (term + termt);
    }
    atomicAdd(lossAcc, local);
}

__global__ void k_final(const float* __restrict__ lossAcc, float* __restrict__ out) {
    if (threadIdx.x == 0 && blockIdx.x == 0) out[0] = lossAcc[0];
}

extern "C" void kernel_launch(void* const* d_in, const int* in_sizes, int n_in,
                              void* d_out, int out_size, void* d_ws, size_t ws_size,
                              hipStream_t stream) {
    const float* modal1 = (const float*)d_in[0];
    const float* modal2 = (const float*)d_in[1];
    const int* targets = (const int*)d_in[2];
    float* out = (float*)d_out;

    char* ws = (char*)d_ws;
    int* counts = (int*)ws;
    float* sumR = (float*)(ws + 256);
    float* sumT = sumR + NCLS * DIMS;
    float* nR = sumT + NCLS * DIMS;
    float* nT = nR + NCLS;
    float* nCc = nT + NCLS;
    float* lossAcc = nCc + NCLS;

    k_zero<<<64, 256, 0, stream>>>((unsigned int*)d_ws);
    k_counts<<<N_SAMP / 256, 256, 0, stream>>>(targets, counts);
    k_sums<<<128, 256, 0, stream>>>(modal1, modal2, targets, sumR, sumT);
    k_norms<<<NCLS, 64, 0, stream>>>(counts, sumR, sumT, nR, nT, nCc);
    k_wmma<<<16, 32, 0, stream>>>(counts, sumR, sumT, nR, nT, nCc, lossAcc);
    k_final<<<1, 1, 0, stream>>>(lossAcc, out);
}